// FPSkNN_50500225466730
// MI455X (gfx1250) — compile-verified
//
#include <hip/hip_runtime.h>

#define BB 4
#define NN 16384
#define DD 256
#define SS 1024
#define KK 32

typedef __attribute__((ext_vector_type(2))) float v2f;
typedef __attribute__((ext_vector_type(8))) float v8f;

// ---------------------------------------------------------------------------
// FPS: one persistent block per batch. 1024 threads, 16 points per thread held
// in registers; running min-dist in registers; argmax via shfl + LDS tree.
// Matches reference: selected[0]=0; s>=1: update dist with selected[s-1],
// selected[s] = first-occurrence argmax.
// ---------------------------------------------------------------------------
__global__ __launch_bounds__(1024) void fps_kernel(const float* __restrict__ xyz,
                                                   int* __restrict__ fps_idx) {
  const int b = blockIdx.x;
  const int tid = threadIdx.x;
  const int lane = tid & 31;
  const int wave = tid >> 5;
  const float* xb = xyz + (size_t)b * NN * 3;

  __shared__ float redV[32];
  __shared__ int   redI[32];
  __shared__ int   sFar;

  float dist[16];
  float px[16], py[16], pz[16];
#pragma unroll
  for (int i = 0; i < 16; ++i) {
    dist[i] = 1e10f;
    int p = tid + (i << 10);
    px[i] = xb[p * 3 + 0];
    py[i] = xb[p * 3 + 1];
    pz[i] = xb[p * 3 + 2];
  }

  int far = 0;
  for (int s = 0; s < SS; ++s) {
    if (tid == 0) fps_idx[b * SS + s] = far;
    if (s == SS - 1) break;

    float cx = xb[far * 3 + 0], cy = xb[far * 3 + 1], cz = xb[far * 3 + 2];
    float bv = -1.0f;
    int bi = 0x7fffffff;
#pragma unroll
    for (int i = 0; i < 16; ++i) {
      float dx = px[i] - cx, dy = py[i] - cy, dz = pz[i] - cz;
      float d = dx * dx + dy * dy + dz * dz;
      float nd = fminf(dist[i], d);
      dist[i] = nd;
      int p = tid + (i << 10);
      if (nd > bv || (nd == bv && p < bi)) { bv = nd; bi = p; }
    }
    // wave-level argmax (tie -> smaller index)
#pragma unroll
    for (int off = 16; off > 0; off >>= 1) {
      float ov = __shfl_xor(bv, off, 32);
      int   oi = __shfl_xor(bi, off, 32);
      if (ov > bv || (ov == bv && oi < bi)) { bv = ov; bi = oi; }
    }
    if (lane == 0) { redV[wave] = bv; redI[wave] = bi; }
    __syncthreads();
    if (wave == 0) {
      float v = redV[lane];
      int   i2 = redI[lane];
#pragma unroll
      for (int off = 16; off > 0; off >>= 1) {
        float ov = __shfl_xor(v, off, 32);
        int   oi = __shfl_xor(i2, off, 32);
        if (ov > v || (ov == v && oi < i2)) { v = ov; i2 = oi; }
      }
      if (lane == 0) sFar = i2;
    }
    __syncthreads();
    far = sFar;
  }
}

// ---------------------------------------------------------------------------
// Gather center_xyz [B,S,3] and center_feature [B,S,256]. One block per (b,s),
// 64 threads, float4 copies.
// ---------------------------------------------------------------------------
__global__ __launch_bounds__(64) void center_gather(const float* __restrict__ xyz,
                                                    const float* __restrict__ feat,
                                                    const int* __restrict__ fps_idx,
                                                    float* __restrict__ center_xyz,
                                                    float* __restrict__ center_feature) {
  const int bs = blockIdx.x;  // b*S + s
  const int b = bs >> 10;
  const int t = threadIdx.x;
  const int idx = fps_idx[bs];
  const float4* src = (const float4*)(feat + ((size_t)b * NN + idx) * DD);
  float4* dst = (float4*)(center_feature + (size_t)bs * DD);
  dst[t] = src[t];
  if (t < 3) center_xyz[(size_t)bs * 3 + t] = xyz[((size_t)b * NN + idx) * 3 + t];
}

// ---------------------------------------------------------------------------
// kNN: block = (b, tile of 16 centers), 256 threads (8 waves).
// Distances via V_WMMA_F32_16X16X4_F32 (K = 3 padded to 4): 16 centers x 16
// points per WMMA; 512-point chunks into LDS; per-wave top-32 merge with
// lexicographic (dist, index) argmin, per-lane used-bitmask (no LDS hazards).
// ---------------------------------------------------------------------------
#define CPAD 516  // row stride (dwords) to break half-wave bank collisions

__global__ __launch_bounds__(256) void knn_kernel(const float* __restrict__ xyz,
                                                  const int* __restrict__ fps_idx,
                                                  int* __restrict__ knn_idx) {
  const int b = blockIdx.x >> 6;
  const int ctile = blockIdx.x & 63;
  const int tid = threadIdx.x;
  const int lane = tid & 31;
  const int wave = tid >> 5;
  const int cl = lane & 15;
  const float* xb = xyz + (size_t)b * NN * 3;

  __shared__ float scx[16], scy[16], scz[16], scs[16];
  __shared__ float cd[16][CPAD];
  __shared__ float topd[16][32];
  __shared__ int   topi[16][32];

  if (tid < 16) {
    int ci = fps_idx[b * SS + ctile * 16 + tid];
    float x = xb[ci * 3 + 0], y = xb[ci * 3 + 1], z = xb[ci * 3 + 2];
    scx[tid] = x; scy[tid] = y; scz[tid] = z;
    scs[tid] = x * x + y * y + z * z;
  }
  for (int i = tid; i < 512; i += 256) {
    ((float*)topd)[i] = 3.0e38f;
    ((int*)topi)[i] = 0;
  }
  __syncthreads();

  // A fragment: 16 centers, K=0..3 (x,y | z,0) per ISA 32-bit A 16x4 layout.
  v2f afrag;
  afrag.x = (lane < 16) ? scx[cl] : scz[cl];
  afrag.y = (lane < 16) ? scy[cl] : 0.0f;

  for (int chunk = 0; chunk < NN; chunk += 512) {
    // ---- distance phase: wave handles point-tiles {wave, wave+8, +16, +24}
#pragma unroll
    for (int tt = 0; tt < 4; ++tt) {
      int t = wave + tt * 8;
      int p = chunk + t * 16 + cl;
      float x = xb[p * 3 + 0], y = xb[p * 3 + 1], z = xb[p * 3 + 2];
      float sqp = x * x + y * y + z * z;
      v2f bfrag;
      bfrag.x = (lane < 16) ? x : z;
      bfrag.y = (lane < 16) ? y : 0.0f;
      v8f acc = {};
      acc = __builtin_amdgcn_wmma_f32_16x16x4_f32(
          /*neg_a=*/false, afrag, /*neg_b=*/false, bfrag,
          /*c_mod=*/(short)0, acc, /*reuse_a=*/false, /*reuse_b=*/false);
      const int mbase = (lane >> 4) << 3;  // lanes 0-15: M=r, lanes 16-31: M=8+r
#pragma unroll
      for (int r = 0; r < 8; ++r) {
        int M = mbase + r;
        cd[M][t * 16 + cl] = scs[M] + sqp - 2.0f * acc[r];
      }
    }
    __syncthreads();

    // ---- merge phase: wave owns centers {wave, wave+8}
#pragma unroll 1
    for (int cc = 0; cc < 2; ++cc) {
      int c = wave + cc * 8;
      unsigned used = 0;            // lane-local consumed slots, m = j>>5
      float keepv = 3.0e38f;
      int keepi = 0;
#pragma unroll 1
      for (int r = 0; r < KK; ++r) {
        float bv = 3.0e38f;
        int bgi = 0x7fffffff;
        int bp = -1;
#pragma unroll 1
        for (int m = 0; m < 17; ++m) {
          if ((used >> m) & 1u) continue;
          int j = lane + (m << 5);  // 0..543
          float v; int gi;
          if (j < 32) { v = topd[c][j]; gi = topi[c][j]; }
          else        { v = cd[c][j - 32]; gi = chunk + (j - 32); }
          if (v < bv || (v == bv && gi < bgi)) { bv = v; bgi = gi; bp = j; }
        }
        // wave all-reduce argmin by (dist, index)
#pragma unroll
        for (int off = 16; off > 0; off >>= 1) {
          float ov = __shfl_xor(bv, off, 32);
          int ogi = __shfl_xor(bgi, off, 32);
          int obp = __shfl_xor(bp, off, 32);
          if (ov < bv || (ov == bv && ogi < bgi)) { bv = ov; bgi = ogi; bp = obp; }
        }
        if ((bp & 31) == lane) used |= 1u << (bp >> 5);
        if (lane == r) { keepv = bv; keepi = bgi; }
      }
      topd[c][lane] = keepv;  // sorted ascending, matches top_k order
      topi[c][lane] = keepi;
    }
    __syncthreads();
  }

  for (int i = tid; i < 512; i += 256) {
    int c = i >> 5, k = i & 31;
    knn_idx[(((size_t)b * SS) + ctile * 16 + c) * KK + k] = topi[c][k];
  }
}

// ---------------------------------------------------------------------------
// Neighbor gather: block per (b,s), 256 threads. neighbor_xyz [B,S,32,3] and
// neighbor_feature [B,S,32,256] (the 128 MB HBM-bound stream).
// ---------------------------------------------------------------------------
__global__ __launch_bounds__(256) void neighbor_gather(const float* __restrict__ xyz,
                                                       const float* __restrict__ feat,
                                                       const int* __restrict__ knn_idx,
                                                       float* __restrict__ nxyz,
                                                       float* __restrict__ nfeat) {
  const int bs = blockIdx.x;
  const int b = bs >> 10;
  const int tid = threadIdx.x;
  __shared__ int idx[32];
  if (tid < 32) idx[tid] = knn_idx[(size_t)bs * KK + tid];
  __syncthreads();

  if (tid < 96) {
    int k = tid / 3, c = tid % 3;
    nxyz[(size_t)bs * 96 + tid] = xyz[((size_t)b * NN + idx[k]) * 3 + c];
  }

  const int k = tid >> 3;   // neighbor 0..31
  const int q = tid & 7;    // 8 threads per neighbor row
  const float4* src = (const float4*)(feat + ((size_t)b * NN + idx[k]) * DD);
  float4* dst = (float4*)(nfeat + ((size_t)bs * KK + k) * DD);
#pragma unroll
  for (int j = 0; j < 8; ++j) {
    dst[j * 8 + q] = src[j * 8 + q];  // 64 float4 per row, coalesced groups
  }
}

// ---------------------------------------------------------------------------
extern "C" void kernel_launch(void* const* d_in, const int* in_sizes, int n_in,
                              void* d_out, int out_size, void* d_ws, size_t ws_size,
                              hipStream_t stream) {
  const float* xyz  = (const float*)d_in[0];   // [4,16384,3]
  const float* feat = (const float*)d_in[1];   // [4,16384,256]
  float* out = (float*)d_out;

  // out layout (flat, return order):
  float* center_xyz     = out;                 // 4*1024*3      = 12288
  float* center_feature = out + 12288;         // 4*1024*256    = 1048576
  float* nxyz           = out + 1060864;       // 4*1024*32*3   = 393216
  float* nfeat          = out + 1454080;       // 4*1024*32*256 = 33554432

  int* fps_idx = (int*)d_ws;                   // 4*1024 ints
  int* knn_idx = fps_idx + BB * SS;            // 4*1024*32 ints (~528 KB total)

  fps_kernel<<<BB, 1024, 0, stream>>>(xyz, fps_idx);
  center_gather<<<BB * SS, 64, 0, stream>>>(xyz, feat, fps_idx, center_xyz, center_feature);
  knn_kernel<<<BB * 64, 256, 0, stream>>>(xyz, fps_idx, knn_idx);
  neighbor_gather<<<BB * SS, 256, 0, stream>>>(xyz, feat, knn_idx, nxyz, nfeat);
}